// GridSampler_4853313044512
// MI455X (gfx1250) — compile-verified
//
#include <hip/hip_runtime.h>
#include <math.h>

// GridSampler: affine grid + bilinear sampling.
// img: (B=16, C=8, H=512, W=512) f32, params: (16,3) f32 -> out same shape.
// Memory-bound gather op: ~256MB traffic -> ~11us floor at 23.3 TB/s HBM.
//
// Round-1 change: per-batch {cos,sin,tx,ty} precomputed by a 16-thread setup
// kernel into d_ws, removing the scalarized Payne-Hanek sinf/cosf bloat
// (375 SALU ops/wave) from the hot kernel. Hot kernel = 4 uniform scalar
// loads + light VALU + 32 gathers + 8 non-temporal stores.

#define B_ 16
#define C_ 8
#define H_ 512
#define W_ 512
#define PLANE (H_ * W_)            // 262144 elements per channel plane

// ---- setup: per-batch affine coefficients -> ws[b] = {cos, sin, tx, ty} ----
__global__ void affine_setup_kernel(const float* __restrict__ params,
                                    float* __restrict__ ws) {
    const int b = threadIdx.x;
    if (b < B_) {
        const float theta = params[b * 3 + 2];
        ws[b * 4 + 0] = cosf(theta);
        ws[b * 4 + 1] = sinf(theta);
        ws[b * 4 + 2] = params[b * 3 + 0] * (1.0f / (float)W_);
        ws[b * 4 + 3] = params[b * 3 + 1] * (1.0f / (float)H_);
    }
}

// ---- shared sampling body ----
__device__ __forceinline__
void sample_tile(const float* __restrict__ img, float* __restrict__ out,
                 int b, float c, float s, float p_tx, float p_ty) {
    // 32x8 output tile per 256-thread block (8 waves of wave32)
    const int tx = threadIdx.x & 31;
    const int ty = threadIdx.x >> 5;
    const int x  = (blockIdx.x << 5) + tx;
    const int y  = (blockIdx.y << 3) + ty;

    // Normalized output coords (match reference exactly)
    const float xs = (2.0f * (float)x + 1.0f) * (1.0f / (float)W_) - 1.0f;
    const float ys = (2.0f * (float)y + 1.0f) * (1.0f / (float)H_) - 1.0f;
    const float gx = fmaf(c, xs, fmaf(-s, ys, p_tx));
    const float gy = fmaf(s, xs, fmaf( c, ys, p_ty));

    // Source pixel-space coords
    const float ix = (fmaf(gx, (float)W_, (float)W_) - 1.0f) * 0.5f;
    const float iy = (fmaf(gy, (float)H_, (float)H_) - 1.0f) * 0.5f;

    const float x0f = floorf(ix);
    const float y0f = floorf(iy);
    const int x0 = (int)x0f;
    const int y0 = (int)y0f;
    const int x1 = x0 + 1;
    const int y1 = y0 + 1;

    const float wx1 = ix - x0f;
    const float wx0 = 1.0f - wx1;
    const float wy1 = iy - y0f;
    const float wy0 = 1.0f - wy1;

    // Per-corner validity folded into weights; clamped indices keep loads in-bounds
    const bool vx0 = (x0 >= 0) & (x0 < W_);
    const bool vx1 = (x1 >= 0) & (x1 < W_);
    const bool vy0 = (y0 >= 0) & (y0 < H_);
    const bool vy1 = (y1 >= 0) & (y1 < H_);

    const float w00 = wy0 * wx0 * (float)(vy0 & vx0);
    const float w01 = wy0 * wx1 * (float)(vy0 & vx1);
    const float w10 = wy1 * wx0 * (float)(vy1 & vx0);
    const float w11 = wy1 * wx1 * (float)(vy1 & vx1);

    const int x0c = min(max(x0, 0), W_ - 1);
    const int x1c = min(max(x1, 0), W_ - 1);
    const int y0c = min(max(y0, 0), H_ - 1);
    const int y1c = min(max(y1, 0), H_ - 1);

    const int o00 = y0c * W_ + x0c;
    const int o01 = y0c * W_ + x1c;
    const int o10 = y1c * W_ + x0c;
    const int o11 = y1c * W_ + x1c;

    const float* __restrict__ base  = img + (size_t)b * C_ * PLANE;
    float* __restrict__       obase = out + (size_t)b * C_ * PLANE + (size_t)(y * W_ + x);

    // Fully unrolled: 32 independent gathers issue back-to-back; channel
    // offsets (ch*1MB <= 7MB) fold into the signed-24-bit IOFFSET so all
    // loads share 4 address VGPR pairs. Stores are non-temporal so the
    // streamed 128MB output doesn't evict the L2-resident input planes.
#pragma unroll
    for (int ch = 0; ch < C_; ++ch) {
        const float* __restrict__ p = base + (size_t)ch * PLANE;
        float v;
        v = w00 * p[o00];
        v = fmaf(w01, p[o01], v);
        v = fmaf(w10, p[o10], v);
        v = fmaf(w11, p[o11], v);
        __builtin_nontemporal_store(v, obase + (size_t)ch * PLANE);
    }
}

// ---- hot kernel: coefficients from workspace (preferred) ----
__global__ __launch_bounds__(256)
void grid_sampler_kernel(const float* __restrict__ img,
                         const float* __restrict__ coef,   // ws: B_ x 4
                         float* __restrict__ out) {
    const int b = blockIdx.z;
    // Uniform per block -> scalar loads (s_load_b128 path, KMcnt)
    const float c    = coef[b * 4 + 0];
    const float s    = coef[b * 4 + 1];
    const float p_tx = coef[b * 4 + 2];
    const float p_ty = coef[b * 4 + 3];
    sample_tile(img, out, b, c, s, p_tx, p_ty);
}

// ---- fallback: trig in-kernel (only if ws too small) ----
__global__ __launch_bounds__(256)
void grid_sampler_kernel_fb(const float* __restrict__ img,
                            const float* __restrict__ params,
                            float* __restrict__ out) {
    const int b = blockIdx.z;
    const float theta = params[b * 3 + 2];
    const float c     = cosf(theta);
    const float s     = sinf(theta);
    const float p_tx  = params[b * 3 + 0] * (1.0f / (float)W_);
    const float p_ty  = params[b * 3 + 1] * (1.0f / (float)H_);
    sample_tile(img, out, b, c, s, p_tx, p_ty);
}

extern "C" void kernel_launch(void* const* d_in, const int* in_sizes, int n_in,
                              void* d_out, int out_size, void* d_ws, size_t ws_size,
                              hipStream_t stream) {
    const float* img    = (const float*)d_in[0];   // 16*8*512*512 f32
    const float* params = (const float*)d_in[1];   // 16*3 f32
    float* out          = (float*)d_out;           // 16*8*512*512 f32

    dim3 grid(W_ / 32, H_ / 8, B_);                // (16, 64, 16) blocks

    if (ws_size >= (size_t)(B_ * 4 * sizeof(float))) {
        float* coef = (float*)d_ws;
        affine_setup_kernel<<<1, B_, 0, stream>>>(params, coef);
        grid_sampler_kernel<<<grid, 256, 0, stream>>>(img, coef, out);
    } else {
        grid_sampler_kernel_fb<<<grid, 256, 0, stream>>>(img, params, out);
    }
}